// FeatureSelect_1580547973607
// MI455X (gfx1250) — compile-verified
//
#include <hip/hip_runtime.h>
#include <hip/hip_bf16.h>
#include <stdint.h>

#define IN_DIM      84
#define SELECT_DIM  42
#define ROW_PAD     85                 // dwords per row in LDS; gcd(85,64)=1 -> conflict-free
#define BLOCK       256
#define CHUNKS_PER_BLOCK  (BLOCK * IN_DIM / 4)        // 5376 float4 chunks per block
#define CHUNKS_PER_THREAD (CHUNKS_PER_BLOCK / BLOCK)  // 21 per thread

__device__ __forceinline__ void async_copy_b128(uint32_t ldsAddr, uint32_t gOff,
                                                const float* __restrict__ base) {
    asm volatile("global_load_async_to_lds_b128 %0, %1, %2 offset:0"
                 :
                 : "v"(ldsAddr), "v"(gOff), "s"(base)
                 : "memory");
}

__device__ __forceinline__ uint32_t chunk_lds_addr(uint32_t ldsBase, uint32_t gl) {
    uint32_t rowL = gl / 21u;                 // local row (0..255)
    uint32_t colC = gl - rowL * 21u;          // float4 column (0..20)
    return ldsBase + rowL * (ROW_PAD * 4u) + colC * 16u;
}

__global__ __launch_bounds__(BLOCK)
void FeatureSelect_kernel(const float* __restrict__ x,
                          float* __restrict__ out,
                          int nrows)
{
    __shared__ float smem[BLOCK * ROW_PAD];   // 256 * 340 B = 87,040 B -> 3 blocks/WGP

    const int tid = threadIdx.x;

    // ---------------------------------------------------------------
    // Stage 256 rows into LDS via CDNA5 async global->LDS copies
    // (ASYNCcnt path), repacking stride-84 -> stride-85 dwords so the
    // uniform-column ds_load_b32 reads below are bank-conflict-free.
    // Consecutive lanes fetch consecutive 16B chunks -> coalesced.
    // Uniform full-block test keeps the hot path branch-free.
    // ---------------------------------------------------------------
    const uint32_t ldsBase = (uint32_t)(uintptr_t)(void*)&smem[0];
    const uint64_t totalChunks = (uint64_t)nrows * (IN_DIM / 4);
    const uint64_t blockChunk0 = (uint64_t)blockIdx.x * CHUNKS_PER_BLOCK;

    if (blockChunk0 + CHUNKS_PER_BLOCK <= totalChunks) {
        #pragma unroll
        for (int c = 0; c < CHUNKS_PER_THREAD; ++c) {
            uint32_t gl = (uint32_t)c * BLOCK + (uint32_t)tid;
            async_copy_b128(chunk_lds_addr(ldsBase, gl),
                            (uint32_t)((blockChunk0 + gl) * 16u), x);
        }
    } else {
        #pragma unroll
        for (int c = 0; c < CHUNKS_PER_THREAD; ++c) {
            uint32_t gl = (uint32_t)c * BLOCK + (uint32_t)tid;
            uint64_t g  = blockChunk0 + gl;
            if (g < totalChunks)
                async_copy_b128(chunk_lds_addr(ldsBase, gl), (uint32_t)(g * 16u), x);
        }
    }
    asm volatile("s_wait_asynccnt 0" ::: "memory");
    __syncthreads();

    // ---------------------------------------------------------------
    // One thread per row. For each of the first 42 elements, count
    // strictly-smaller elements with FOUR independent cmp/add chains
    // (ILP for dual-issue), then set the first CLEAR bit >= cnt in a
    // 96-bit mask. Stable sort places equal values at consecutive
    // ranks in index order and k ascends, so "first clear bit >= cnt"
    // reproduces the stable ranking exactly with no per-pair tie work.
    // ---------------------------------------------------------------
    const int row = blockIdx.x * BLOCK + tid;
    if (row < nrows) {
        const float* srow = &smem[tid * ROW_PAD];

        float xr[IN_DIM];                 // static indices only -> VGPRs
        #pragma unroll
        for (int m = 0; m < IN_DIM; ++m) xr[m] = srow[m];

        uint32_t m0 = 0u, m1 = 0u, m2 = 0u;

        #pragma unroll 1   // bound code size; 84-wide inner loop is unrolled
        for (int k = 0; k < SELECT_DIM; ++k) {
            const float xk = srow[k];     // uniform col -> conflict-free ds_load_b32

            // 4 independent accumulator chains over the 84 registers
            int c0 = 0, c1 = 0, c2 = 0, c3 = 0;
            #pragma unroll
            for (int m = 0; m < IN_DIM; m += 4) {
                c0 += (int)(xr[m + 0] < xk);
                c1 += (int)(xr[m + 1] < xk);
                c2 += (int)(xr[m + 2] < xk);
                c3 += (int)(xr[m + 3] < xk);
            }
            const uint32_t r = (uint32_t)((c0 + c1) + (c2 + c3));

            // insert at first clear bit >= r of (m2:m1:m0)
            const uint32_t f  = (1u << (r & 31u)) - 1u;       // partial low mask
            const uint32_t l0 = (r < 32u) ? f : 0xFFFFFFFFu;
            const uint32_t l1 = (r < 32u) ? 0u : (r < 64u) ? f : 0xFFFFFFFFu;
            const uint32_t l2 = (r < 64u) ? 0u : f;
            const uint32_t x0 = ~(m0 | l0);                   // clear-bit candidates
            const uint32_t x1 = ~(m1 | l1);
            const uint32_t x2 = ~(m2 | l2);
            const uint32_t b0 = x0 & (0u - x0);               // isolate lowest set bit
            const uint32_t b1 = x1 & (0u - x1);
            const uint32_t b2 = x2 & (0u - x2);
            const bool z0  = (x0 == 0u);
            const bool z01 = z0 & (x1 == 0u);
            m0 |= b0;
            m1 |= z0  ? b1 : 0u;
            m2 |= z01 ? b2 : 0u;
        }

        // Expand bitmask -> 84 floats, 21 coalesced b128 stores.
        auto bitf = [&](int i) -> float {
            const uint32_t w = (i < 32) ? m0 : (i < 64) ? m1 : m2;
            return ((w >> (i & 31)) & 1u) ? 1.0f : 0.0f;
        };
        float4* orow = (float4*)(out + (size_t)row * IN_DIM);
        #pragma unroll
        for (int j = 0; j < IN_DIM / 4; ++j) {
            float4 f;
            f.x = bitf(j * 4 + 0);
            f.y = bitf(j * 4 + 1);
            f.z = bitf(j * 4 + 2);
            f.w = bitf(j * 4 + 3);
            orow[j] = f;
        }
    }
}

extern "C" void kernel_launch(void* const* d_in, const int* in_sizes, int n_in,
                              void* d_out, int out_size, void* d_ws, size_t ws_size,
                              hipStream_t stream) {
    const float* x = (const float*)d_in[0];
    float* out = (float*)d_out;
    const int nrows  = in_sizes[0] / IN_DIM;    // 1,000,000
    const int blocks = (nrows + BLOCK - 1) / BLOCK;
    hipLaunchKernelGGL(FeatureSelect_kernel, dim3(blocks), dim3(BLOCK), 0, stream,
                       x, out, nrows);
}